// Downsample_12240656793719
// MI455X (gfx1250) — compile-verified
//
#include <hip/hip_runtime.h>
#include <stdint.h>

// Sparse-conv 2x downsample: out[m,c] = mean_k feats[results[k,m], c], with
// results==-1 contributing a zero row.  Bandwidth-bound (~0.4 FLOP/B).
// CDNA5 strategy: double-buffered ASYNCcnt gather pipeline
// (global_load_async_to_lds_b128, GVS addressing), NT hints on the streaming
// index/output traffic so the 192MB L2 is reserved for the 256MB feats table.

typedef float v4f __attribute__((ext_vector_type(4)));

#define KK 8              // kernel offsets (2x2x2)
#define CC 64             // channels
#define WPB 4             // waves per block (128 threads, wave32)
#define ITERS 8           // target pipeline iterations per wave
#define ROW_BYTES 512     // one k-slot: 2 voxels x 64ch x 4B, 16B per lane
#define STAGE_BYTES (KK * ROW_BYTES)  // 4KB per stage

// Issue one pipeline stage for a voxel pair: 8 async b128 gathers, each moving
// two full 256B feature rows (lanes 0-15 -> voxel 2p, lanes 16-31 -> 2p+1).
// Indices stay in VGPRs for the consume-side zero-row mask; idx==-1 loads are
// clamped to row 0 and weighted out.
__device__ __forceinline__ void issue_stage(uint64_t feats_base,
                                            const int* __restrict__ results,
                                            int M, int mcl, int lane4b,
                                            uint32_t lds_base, int* idxs) {
#pragma unroll
  for (int k = 0; k < KK; ++k) {
    const int idx = __builtin_nontemporal_load(results + (size_t)k * M + mcl);
    idxs[k] = idx;
    const int cidx = idx < 0 ? 0 : idx;
    const uint32_t voff = ((uint32_t)cidx << 8) + (uint32_t)lane4b; // idx*256 + lane*16
    const uint32_t lds = lds_base + (uint32_t)(k * ROW_BYTES);
    // GVS form: 64-bit SGPR base + 32-bit per-lane offset.
    asm volatile("global_load_async_to_lds_b128 %0, %1, %2"
                 :: "v"(lds), "v"(voff), "s"(feats_base)
                 : "memory");
  }
}

__global__ __launch_bounds__(WPB * 32)
void Downsample_12240656793719_kernel(const float* __restrict__ feats,
                                      const int* __restrict__ results,
                                      float* __restrict__ out, int M) {
  // 2 stages x 4KB per wave; 4 waves -> 32KB static LDS.
  __shared__ __align__(16) float smem[WPB * 2 * STAGE_BYTES / 4];

  const int lane = threadIdx.x & 31;
  const int wave = threadIdx.x >> 5;
  const int npairs = (M + 1) >> 1;
  const int stride = gridDim.x * WPB;
  int p = blockIdx.x * WPB + wave;     // one voxel pair per wave-iteration
  if (p >= npairs) return;

  const int half = lane >> 4;          // 0: voxel 2p, 1: voxel 2p+1
  const int sub = lane & 15;           // 16B channel slice within the row
  const int lane4b = lane << 4;        // byte offset of lane's slice in a 512B k-slot

  float* stA = smem + wave * (2 * STAGE_BYTES / 4);
  float* stB = stA + STAGE_BYTES / 4;
  uint32_t ldsA = (uint32_t)(uintptr_t)stA + (uint32_t)lane4b; // LDS byte offset
  uint32_t ldsB = ldsA + STAGE_BYTES;
  const uint64_t feats_base = (uint64_t)(uintptr_t)feats;

  int m = 2 * p + half;
  int mcl = m < M ? m : (M - 1);       // odd-M tail: clamp loads, mask store

  int idxA[KK], idxB[KK];
  issue_stage(feats_base, results, M, mcl, lane4b, ldsA, idxA);

  for (;;) {
    const int pn = p + stride;
    const bool has_next = (pn < npairs);
    int mn = 2 * pn + half;
    if (has_next) {
      const int mncl = mn < M ? mn : (M - 1);
      issue_stage(feats_base, results, M, mncl, lane4b, ldsB, idxB);
      // Two stages (<=16 ops) in flight; async loads complete in order, so
      // ASYNCcnt<=8 means stage A has landed in LDS.
      asm volatile("s_wait_asynccnt 8" ::: "memory");
    } else {
      asm volatile("s_wait_asynccnt 0" ::: "memory");
    }

    // Consume stage A: weighted sum over the 8 kernel offsets.  w=0.125 is a
    // power of two, so per-term scaling is exact (bit-identical to
    // sum-then-divide in the same order); w=0 realizes the zero row.
    v4f acc = {0.f, 0.f, 0.f, 0.f};
#pragma unroll
    for (int k = 0; k < KK; ++k) {
      const v4f v = *(const v4f*)(stA + k * (ROW_BYTES / 4) + lane * 4);
      const float wk = idxA[k] >= 0 ? 0.125f : 0.0f;
      acc += v * wk;
    }
    if (m < M) {
      // Streaming output: keep it out of L2 so feats lines stay resident.
      __builtin_nontemporal_store(acc, (v4f*)(out + (size_t)m * CC + sub * 4));
    }

    if (!has_next) break;
    p = pn;
    m = mn;
    float* tf = stA; stA = stB; stB = tf;
    uint32_t tu = ldsA; ldsA = ldsB; ldsB = tu;
#pragma unroll
    for (int k = 0; k < KK; ++k) idxA[k] = idxB[k];
  }
}

extern "C" void kernel_launch(void* const* d_in, const int* in_sizes, int n_in,
                              void* d_out, int out_size, void* d_ws, size_t ws_size,
                              hipStream_t stream) {
  const float* feats   = (const float*)d_in[0];  // [N, 64] f32
  const int*   results = (const int*)d_in[1];    // [8, M] i32
  float*       out     = (float*)d_out;          // [M, 64] f32
  (void)d_ws; (void)ws_size; (void)n_in; (void)out_size;

  const int M = in_sizes[1] / KK;
  const int npairs = (M + 1) >> 1;
  int waves_needed = (npairs + ITERS - 1) / ITERS;
  int blocks = (waves_needed + WPB - 1) / WPB;
  if (blocks < 1) blocks = 1;

  Downsample_12240656793719_kernel<<<blocks, WPB * 32, 0, stream>>>(
      feats, results, out, M);
}